// FinalDistributionLayer_25795573579999
// MI455X (gfx1250) — compile-verified
//
#include <hip/hip_runtime.h>
#include <hip/hip_bf16.h>
#include <stdint.h>

// Pointer-generator final distribution for MI455X (gfx1250, wave32).
//
// out[t,b,v]        = p_gen[t,b] * vocab_dists[t,b,v]         (v < V)
// out[t,b,V:V+OOV]  = 0
// out[t,b,id[b,l]] += (1 - p_gen[t,b]) * attn[t,b,l]          (scatter-add, dup accumulate)
//
// Kernel 1: streaming gate+pad, float4 (global_load_b128 NT / global_store_b128).
// Kernel 2: tiny scatter via hardware global_atomic_add_f32.

typedef __attribute__((ext_vector_type(4))) float v4f;

// ---------------------------------------------------------------------------
// Kernel 1 (vectorized): one float4 column chunk per thread.
// grid.y = t*B + b  (row), grid.x tiles EV/4 columns.
// vocab_dists is streamed once -> non-temporal load (th:NT) keeps L2 free for
// the 103MB output, which fits in the 192MB L2 and is re-touched by kernel 2.
// ---------------------------------------------------------------------------
__global__ void __launch_bounds__(256)
pg_gate_pad_vec4(const v4f* __restrict__ vocab,   // [TB, V/4]
                 const float* __restrict__ pgen,  // [TB]
                 v4f* __restrict__ out,           // [TB, EV/4]
                 int V4, int EV4) {
    const int row = blockIdx.y;                       // t*B + b
    const int c4  = blockIdx.x * blockDim.x + threadIdx.x;
    if (c4 >= EV4) return;

    const float p = pgen[row];                        // wave-uniform address
    v4f r = (v4f){0.0f, 0.0f, 0.0f, 0.0f};
    if (c4 < V4) {
        const v4f v = __builtin_nontemporal_load(&vocab[(size_t)row * V4 + c4]);
        r.x = p * v.x;
        r.y = p * v.y;
        r.z = p * v.z;
        r.w = p * v.w;
    }
    out[(size_t)row * EV4 + c4] = r;                  // RT store: stays in L2
}

// Scalar fallback (only used if V or EV is not a multiple of 4).
__global__ void __launch_bounds__(256)
pg_gate_pad_scalar(const float* __restrict__ vocab,
                   const float* __restrict__ pgen,
                   float* __restrict__ out,
                   int V, int EV) {
    const int row = blockIdx.y;
    const int c   = blockIdx.x * blockDim.x + threadIdx.x;
    if (c >= EV) return;
    const float p = pgen[row];
    float r = 0.0f;
    if (c < V) r = p * __builtin_nontemporal_load(&vocab[(size_t)row * V + c]);
    out[(size_t)row * EV + c] = r;
}

// ---------------------------------------------------------------------------
// Kernel 2: scatter-add of copy distribution. T*B*L threads total (~205K):
// negligible next to the stream; duplicates require atomics -> hardware
// global_atomic_add_f32 (relaxed, agent scope), hitting L2-resident output.
// ---------------------------------------------------------------------------
__global__ void __launch_bounds__(256)
pg_scatter_add(const float* __restrict__ attn,   // [T,B,L] flat = [tb, l]
               const float* __restrict__ pgen,   // [TB]
               const int*   __restrict__ ids,    // [B, L]
               float* __restrict__ out,          // [TB, EV]
               int B, int L, int EV, int total) {
    const int idx = blockIdx.x * blockDim.x + threadIdx.x;
    if (idx >= total) return;

    const int l  = idx % L;
    const int tb = idx / L;                       // t*B + b
    const int b  = tb % B;

    const float p   = pgen[tb];
    const float val = (1.0f - p) * attn[idx];
    const int   id  = ids[b * L + l];

    if ((unsigned)id < (unsigned)EV) {
        __hip_atomic_fetch_add(&out[(size_t)tb * EV + id], val,
                               __ATOMIC_RELAXED, __HIP_MEMORY_SCOPE_AGENT);
    }
}

// ---------------------------------------------------------------------------
// Host launcher. Shapes recovered from in_sizes:
//   in_sizes[0] = T*B*V, in_sizes[1] = T*B*L, in_sizes[2] = T*B,
//   in_sizes[3] = B*L,   out_size    = T*B*EV
// ---------------------------------------------------------------------------
extern "C" void kernel_launch(void* const* d_in, const int* in_sizes, int n_in,
                              void* d_out, int out_size, void* d_ws, size_t ws_size,
                              hipStream_t stream) {
    const float* vocab = (const float*)d_in[0];   // [T,B,V]
    const float* attn  = (const float*)d_in[1];   // [T,B,L]
    const float* pgen  = (const float*)d_in[2];   // [T,B,1]
    const int*   ids   = (const int*)d_in[3];     // [B,L]
    float*       out   = (float*)d_out;           // [T,B,EV]

    const int TB = in_sizes[2];                   // T*B  (512)
    const int L  = in_sizes[1] / TB;              // 400
    const int B  = in_sizes[3] / L;               // 16
    const int V  = in_sizes[0] / TB;              // 50000
    const int EV = out_size / TB;                 // 50100

    const int threads = 256;                      // 8 wave32 per block

    if ((V % 4 == 0) && (EV % 4 == 0)) {
        const int V4  = V / 4;
        const int EV4 = EV / 4;
        dim3 grid((EV4 + threads - 1) / threads, TB);
        pg_gate_pad_vec4<<<grid, threads, 0, stream>>>(
            (const v4f*)vocab, pgen, (v4f*)out, V4, EV4);
    } else {
        dim3 grid((EV + threads - 1) / threads, TB);
        pg_gate_pad_scalar<<<grid, threads, 0, stream>>>(vocab, pgen, out, V, EV);
    }

    const int total = TB * L;                     // 204,800
    pg_scatter_add<<<(total + threads - 1) / threads, threads, 0, stream>>>(
        attn, pgen, ids, out, B, L, EV, total);
}